// GRUNet_17317308137768
// MI455X (gfx1250) — compile-verified
//
#include <hip/hip_runtime.h>

// GRU (B=8192, T=512, I=1, H=8) + FC, fully fused, wave32.
// Recurrent GEMM per timestep via V_WMMA_F32_16X16X4_F32:
//   A (16x4 f32, loop-invariant) = W_hh tile (gates as M, hidden as K)
//   B (4x16 f32)                 = h^T       (batch as N, 16 seqs/wave)
//   C (16x16 f32)                = biases folded in
// r and z sigmoids computed in one full-wave pass (r: lanes 0-15,
// z: lanes 16-31); half-wave swaps via v_permlanex16 (VALU, no LDS).

typedef float v2f __attribute__((ext_vector_type(2)));
typedef float v8f __attribute__((ext_vector_type(8)));

#define T_LEN 512
#define H_DIM 8

static __device__ __forceinline__ float fast_sigmoid(float t) {
  // 1 / (1 + exp(-t)) = 1 / (1 + exp2(-t * log2 e))
  float e = __builtin_amdgcn_exp2f(t * -1.4426950408889634f);
  return __builtin_amdgcn_rcpf(1.0f + e);
}

static __device__ __forceinline__ float fast_tanh(float t) {
  // tanh(t) = 1 - 2 / (exp2(2 t log2 e) + 1)
  float e = __builtin_amdgcn_exp2f(t * 2.8853900817779268f);
  return fmaf(-2.0f, __builtin_amdgcn_rcpf(e + 1.0f), 1.0f);
}

// Exact lane(i) <-> lane(i^16) swap, VALU only (v_permlanex16_b32).
static __device__ __forceinline__ float xhalf(float v) {
  int i = __builtin_bit_cast(int, v);
  int r = __builtin_amdgcn_permlanex16(i, i, 0x76543210, 0xfedcba98, false, false);
  return __builtin_bit_cast(float, r);
}

__global__ __launch_bounds__(256)
void gru_wmma16_kernel(const float* __restrict__ x,
                       const float* __restrict__ W_ih,
                       const float* __restrict__ W_hh,
                       const float* __restrict__ b_ih,
                       const float* __restrict__ b_hh,
                       const float* __restrict__ W_fc,
                       const float* __restrict__ b_fc,
                       float* __restrict__ out)
{
  const int lane   = threadIdx.x & 31;
  const int lb     = lane & 15;     // batch sub-index within wave tile
  const int hiHalf = lane >> 4;     // 0 = lanes 0-15, 1 = lanes 16-31
  const int wave   = blockIdx.x * (blockDim.x >> 5) + (threadIdx.x >> 5);
  const int batch  = wave * 16 + lb;

  // ---- A operand: W_hh [24][8], gates as M ----------------------------
  // 32-bit A 16x4 layout: VGPR0 = K {0 | 2}, VGPR1 = K {1 | 3} (low|high half)
  const int colb = hiHalf * 2;
  v2f a0c0, a0c1, a1c0, a1c1;
  a0c0.x = W_hh[lb * H_DIM + colb + 0];
  a0c0.y = W_hh[lb * H_DIM + colb + 1];
  a0c1.x = W_hh[lb * H_DIM + 4 + colb + 0];
  a0c1.y = W_hh[lb * H_DIM + 4 + colb + 1];
  const bool row1ok = (lb < 8);           // tile1 rows 8..15 are padding
  const int r1 = 16 + lb;
  a1c0.x = row1ok ? W_hh[r1 * H_DIM + colb + 0] : 0.0f;
  a1c0.y = row1ok ? W_hh[r1 * H_DIM + colb + 1] : 0.0f;
  a1c1.x = row1ok ? W_hh[r1 * H_DIM + 4 + colb + 0] : 0.0f;
  a1c1.y = row1ok ? W_hh[r1 * H_DIM + 4 + colb + 1] : 0.0f;

  // ---- C operand: biases folded into the accumulator ------------------
  // tile0: pre_r (low half) / pre_z (high half) get b_ih+b_hh
  // tile1: pre_n gets only b_hh (b_ih_n added outside r*hn)
  v8f c0, c1;
#pragma unroll
  for (int r = 0; r < 8; ++r) {
    int g0 = r + 8 * hiHalf;
    c0[r] = b_ih[g0] + b_hh[g0];
    c1[r] = hiHalf ? 0.0f : b_hh[16 + r];
  }

  // ---- Uniform scalar weights (SGPRs) ---------------------------------
  float s_win[8], s_bin[8], s_wfc[8];
  float w01[8];   // per-lane: W_ih_r (low half) / W_ih_z (high half)
#pragma unroll
  for (int k = 0; k < 8; ++k) {
    w01[k]   = hiHalf ? W_ih[8 + k] : W_ih[k];
    s_win[k] = W_ih[16 + k];
    s_bin[k] = b_ih[16 + k];
    s_wfc[k] = W_fc[k];
  }
  const float s_bfc = b_fc[0];

  // ---- State ----------------------------------------------------------
  // B operand layout (4x16): VGPR0 = K {0 | 2}, VGPR1 = K {1 | 3}
  v2f bb0 = {0.0f, 0.0f};   // K=0..3 chunk of h^T
  v2f bb1 = {0.0f, 0.0f};   // K=4..7 chunk of h^T
  float h[8];
#pragma unroll
  for (int k = 0; k < 8; ++k) h[k] = 0.0f;

  const float4* xrow = (const float4*)(x + (size_t)batch * T_LEN);
  float4* orow4 = (float4*)(out + (size_t)batch * T_LEN);

  float4 xq = xrow[0];                 // software-pipelined x stream
  for (int t4 = 0; t4 < T_LEN / 4; ++t4) {
    float4 xq_next = xrow[(t4 + 1) & (T_LEN / 4 - 1)];  // prefetch (wraps harmlessly)
    float xs[4] = {xq.x, xq.y, xq.z, xq.w};
    float ybuf[4];

#pragma unroll
    for (int s = 0; s < 4; ++s) {
      float xv = xs[s];

      // hg = W_hh @ h + biases : 2 gate tiles x 2 K-chunks
      v8f d0 = __builtin_amdgcn_wmma_f32_16x16x4_f32(
          false, a0c0, false, bb0, (short)0, c0, false, false);
      d0 = __builtin_amdgcn_wmma_f32_16x16x4_f32(
          false, a0c1, false, bb1, (short)0, d0, false, false);
      v8f d1 = __builtin_amdgcn_wmma_f32_16x16x4_f32(
          false, a1c0, false, bb0, (short)0, c1, false, false);
      v8f d1f = __builtin_amdgcn_wmma_f32_16x16x4_f32(
          false, a1c1, false, bb1, (short)0, d1, false, false);

      // One full-wave sigmoid pass: r in lanes 0-15, z in lanes 16-31.
      float g[8];
#pragma unroll
      for (int k = 0; k < 8; ++k)
        g[k] = fast_sigmoid(fmaf(xv, w01[k], d0[k]));

      // n-gate, state update, fused FC (valid in lanes 0-15).
      float y = s_bfc;
#pragma unroll
      for (int k = 0; k < 8; ++k) {
        float zg = xhalf(g[k]);                        // z from upper half
        float xn = fmaf(xv, s_win[k], s_bin[k]);
        float ng = fast_tanh(fmaf(g[k], d1f[k], xn));  // g[k]=r in low half
        float hk = fmaf(zg, h[k] - ng, ng);            // (1-z)*n + z*h
        h[k] = hk;
        y = fmaf(hk, s_wfc[k], y);
      }
      ybuf[s] = y;

      // Rebuild B operand: high half needs h2,h3 / h6,h7.
      float u2 = xhalf(h[2]);
      float u3 = xhalf(h[3]);
      float u6 = xhalf(h[6]);
      float u7 = xhalf(h[7]);
      bb0.x = hiHalf ? u2 : h[0];
      bb0.y = hiHalf ? u3 : h[1];
      bb1.x = hiHalf ? u6 : h[4];
      bb1.y = hiHalf ? u7 : h[5];
    }

    if (lane < 16)
      orow4[t4] = make_float4(ybuf[0], ybuf[1], ybuf[2], ybuf[3]);
    xq = xq_next;
  }
}

extern "C" void kernel_launch(void* const* d_in, const int* in_sizes, int n_in,
                              void* d_out, int out_size, void* d_ws, size_t ws_size,
                              hipStream_t stream) {
  const float* x    = (const float*)d_in[0];
  const float* W_ih = (const float*)d_in[1];
  const float* W_hh = (const float*)d_in[2];
  const float* b_ih = (const float*)d_in[3];
  const float* b_hh = (const float*)d_in[4];
  const float* W_fc = (const float*)d_in[5];
  const float* b_fc = (const float*)d_in[6];
  float* out = (float*)d_out;

  const int B = in_sizes[0] / T_LEN;   // 8192
  const int wavesTotal = B / 16;       // 512 waves, 16 sequences each
  const int wavesPerBlock = 8;         // 256 threads/block
  dim3 grid(wavesTotal / wavesPerBlock);
  dim3 block(wavesPerBlock * 32);
  hipLaunchKernelGGL(gru_wmma16_kernel, grid, block, 0, stream,
                     x, W_ih, W_hh, b_ih, b_hh, W_fc, b_fc, out);
}